// TransferEntropyEstimator_23270132810167
// MI455X (gfx1250) — compile-verified
//
#include <hip/hip_runtime.h>

// ---------------- constants ----------------
#define K_BINS   8
#define HIST_K   3
#define TLEN     8388608                 // 2^23
#define NSAMP    (TLEN - HIST_K - 1)     // 8388604
#define NPAST    512                     // 8^3
#define CH       4096                    // samples per staged chunk
#define BDIM     256
#define HIST_BLOCKS 512                  // persistent blocks: 4 chunks each

#define AS1 __attribute__((address_space(1)))
#define AS3 __attribute__((address_space(3)))

typedef int v4i __attribute__((vector_size(16)));

#if defined(__has_builtin)
# if __has_builtin(__builtin_amdgcn_global_load_async_to_lds_b128)
#  define USE_ASYNC_LDS 1
# endif
#endif
#ifndef USE_ASYNC_LDS
# define USE_ASYNC_LDS 0
#endif

// ordered-uint encoding so atomicMin/Max on unsigned give float min/max
__device__ __forceinline__ unsigned enc_f(float f) {
  unsigned u = __float_as_uint(f);
  return (u & 0x80000000u) ? ~u : (u | 0x80000000u);
}
__device__ __forceinline__ float dec_f(unsigned u) {
  unsigned b = (u & 0x80000000u) ? (u ^ 0x80000000u) : ~u;
  return __uint_as_float(b);
}

// ---------------- init ----------------
__global__ void k_init(unsigned* ctrl) {
  if (blockIdx.x == 0 && threadIdx.x == 0) {
    ctrl[0] = 0xFFFFFFFFu;  // src min (encoded)
    ctrl[1] = 0u;           // src max
    ctrl[2] = 0xFFFFFFFFu;  // tgt min
    ctrl[3] = 0u;           // tgt max
    float* acc = (float*)(ctrl + 4);
    acc[0] = 0.f; acc[1] = 0.f; acc[2] = 0.f; acc[3] = 0.f;
  }
}

// ---------------- fused min/max of both series ----------------
__global__ void __launch_bounds__(BDIM)
k_minmax(const float4* __restrict__ s, const float4* __restrict__ t,
         unsigned* ctrl, int nvec) {
  unsigned smn = 0xFFFFFFFFu, smx = 0u, tmn = 0xFFFFFFFFu, tmx = 0u;
  for (int i = blockIdx.x * blockDim.x + threadIdx.x; i < nvec;
       i += gridDim.x * blockDim.x) {
    float4 a = s[i], b = t[i];
    unsigned e;
    e = enc_f(a.x); smn = e < smn ? e : smn; smx = e > smx ? e : smx;
    e = enc_f(a.y); smn = e < smn ? e : smn; smx = e > smx ? e : smx;
    e = enc_f(a.z); smn = e < smn ? e : smn; smx = e > smx ? e : smx;
    e = enc_f(a.w); smn = e < smn ? e : smn; smx = e > smx ? e : smx;
    e = enc_f(b.x); tmn = e < tmn ? e : tmn; tmx = e > tmx ? e : tmx;
    e = enc_f(b.y); tmn = e < tmn ? e : tmn; tmx = e > tmx ? e : tmx;
    e = enc_f(b.z); tmn = e < tmn ? e : tmn; tmx = e > tmx ? e : tmx;
    e = enc_f(b.w); tmn = e < tmn ? e : tmn; tmx = e > tmx ? e : tmx;
  }
  __shared__ unsigned red[4][BDIM];
  int tid = threadIdx.x;
  red[0][tid] = smn; red[1][tid] = smx; red[2][tid] = tmn; red[3][tid] = tmx;
  __syncthreads();
  for (int off = BDIM / 2; off > 0; off >>= 1) {
    if (tid < off) {
      unsigned v;
      v = red[0][tid + off]; if (v < red[0][tid]) red[0][tid] = v;
      v = red[1][tid + off]; if (v > red[1][tid]) red[1][tid] = v;
      v = red[2][tid + off]; if (v < red[2][tid]) red[2][tid] = v;
      v = red[3][tid + off]; if (v > red[3][tid]) red[3][tid] = v;
    }
    __syncthreads();
  }
  if (tid == 0) {
    atomicMin(&ctrl[0], red[0][0]);
    atomicMax(&ctrl[1], red[1][0]);
    atomicMin(&ctrl[2], red[2][0]);
    atomicMax(&ctrl[3], red[3][0]);
  }
}

// ---------------- discretize both series to u8 (matches reference formula) ----
__device__ __forceinline__ unsigned char binv(float v, float mn, float denom, int flat) {
  float nrm = (v - mn) / denom;
  int b = (int)(nrm * (float)(K_BINS - 1));  // trunc toward zero like astype(int32)
  b = b < 0 ? 0 : (b > (K_BINS - 1) ? (K_BINS - 1) : b);
  return flat ? (unsigned char)0 : (unsigned char)b;
}

__global__ void __launch_bounds__(BDIM)
k_disc(const float4* __restrict__ s, const float4* __restrict__ t,
       uchar4* __restrict__ s8, uchar4* __restrict__ t8,
       const unsigned* ctrl, int nvec) {
  float smn = dec_f(ctrl[0]), smx = dec_f(ctrl[1]);
  float tmn = dec_f(ctrl[2]), tmx = dec_f(ctrl[3]);
  float srng = smx - smn, trng = tmx - tmn;
  float sden = srng + 1e-8f, tden = trng + 1e-8f;
  int sflat = srng < 1e-8f, tflat = trng < 1e-8f;
  for (int i = blockIdx.x * blockDim.x + threadIdx.x; i < nvec;
       i += gridDim.x * blockDim.x) {
    float4 a = s[i], b = t[i];
    uchar4 oa, ob;
    oa.x = binv(a.x, smn, sden, sflat);
    oa.y = binv(a.y, smn, sden, sflat);
    oa.z = binv(a.z, smn, sden, sflat);
    oa.w = binv(a.w, smn, sden, sflat);
    ob.x = binv(b.x, tmn, tden, tflat);
    ob.y = binv(b.y, tmn, tden, tflat);
    ob.z = binv(b.z, tmn, tden, tflat);
    ob.w = binv(b.w, tmn, tden, tflat);
    s8[i] = oa;
    t8[i] = ob;
  }
}

// ---------------- histogram build (both directions in one pass) ----------------
__global__ void __launch_bounds__(BDIM)
k_hist(const unsigned char* __restrict__ s8, const unsigned char* __restrict__ t8,
       unsigned* __restrict__ jy_xy, unsigned* __restrict__ jy_yx,
       unsigned* __restrict__ jc_xy, unsigned* __restrict__ jc_yx, int n) {
  __shared__ unsigned shY[2 * NPAST * K_BINS];           // 32 KB: two 512x8 hists
  __shared__ __align__(16) unsigned char tS[CH + 32];    // staged source bytes
  __shared__ __align__(16) unsigned char tT[CH + 32];    // staged target bytes

  for (int i = threadIdx.x; i < 2 * NPAST * K_BINS; i += blockDim.x) shY[i] = 0u;
  __syncthreads();

  for (int c0 = blockIdx.x * CH; c0 < n; c0 += gridDim.x * CH) {
    int nb = min(CH + 16, TLEN - c0);   // bytes needed from [c0, ...)
    int nu = (nb + 15) >> 4;            // 16-byte units
#if USE_ASYNC_LDS
    for (int u = threadIdx.x; u < nu; u += blockDim.x) {
      __builtin_amdgcn_global_load_async_to_lds_b128(
          (AS1 v4i*)(s8 + c0 + 16 * u), (AS3 v4i*)(tS + 16 * u), 0, 0);
      __builtin_amdgcn_global_load_async_to_lds_b128(
          (AS1 v4i*)(t8 + c0 + 16 * u), (AS3 v4i*)(tT + 16 * u), 0, 0);
    }
# if __has_builtin(__builtin_amdgcn_s_wait_asynccnt)
    __builtin_amdgcn_s_wait_asynccnt(0);
# else
    asm volatile("s_wait_asynccnt 0" ::: "memory");
# endif
#else
    for (int u = threadIdx.x; u < nu; u += blockDim.x) {
      ((uint4*)tS)[u] = ((const uint4*)(s8 + c0))[u];
      ((uint4*)tT)[u] = ((const uint4*)(t8 + c0))[u];
    }
#endif
    __syncthreads();

    int jend = min(CH, n - c0);
    for (int j = threadIdx.x; j < jend; j += blockDim.x) {
      unsigned t1 = tT[j + 1], t2 = tT[j + 2], t3 = tT[j + 3], yb = tT[j + 4];
      unsigned s1 = tS[j + 1], s2 = tS[j + 2], s3 = tS[j + 3], xb = tS[j + 4];
      unsigned yp = t3 + 8u * t2 + 64u * t1;     // y_past_hash
      unsigned xp = s3 + 8u * s2 + 64u * s1;     // x_past_hash
      atomicAdd(&shY[yp * 8u + yb], 1u);                         // joint_y (x->y)
      atomicAdd(&shY[NPAST * K_BINS + xp * 8u + xb], 1u);        // joint_y (y->x)
      atomicAdd(&jc_xy[((yp << 9) | xp) * 8u + yb], 1u);         // joint_c (x->y)
      atomicAdd(&jc_yx[((xp << 9) | yp) * 8u + xb], 1u);         // joint_c (y->x)
    }
    __syncthreads();   // tiles reused next chunk
  }

  // one flush per persistent block (HIST_BLOCKS blocks total, not one per chunk)
  for (int i = threadIdx.x; i < NPAST * K_BINS; i += blockDim.x) {
    unsigned v = shY[i];                 if (v) atomicAdd(&jy_xy[i], v);
    v = shY[NPAST * K_BINS + i];         if (v) atomicAdd(&jy_yx[i], v);
  }
}

// ---------------- conditional-entropy reduction ----------------
__device__ __forceinline__ float plg(unsigned c) {
  if (c == 0u) return 0.f;
  float f = (float)c;
  return f * log2f(f);
}

__global__ void __launch_bounds__(BDIM)
k_entropy(const uint4* __restrict__ counts, int rows, float* acc) {
  float loc = 0.f;
  for (int r = blockIdx.x * blockDim.x + threadIdx.x; r < rows;
       r += gridDim.x * blockDim.x) {
    uint4 a = counts[2 * r], b = counts[2 * r + 1];
    unsigned tot = a.x + a.y + a.z + a.w + b.x + b.y + b.z + b.w;
    if (tot >= 2u) {
      float s = plg(a.x) + plg(a.y) + plg(a.z) + plg(a.w)
              + plg(b.x) + plg(b.y) + plg(b.z) + plg(b.w);
      float ft = (float)tot;
      // weight*ent = (1/n) * (tot*log2(tot) - sum ci*log2(ci)); /n applied later
      loc += ft * log2f(ft) - s;
    }
  }
  __shared__ float red[BDIM];
  red[threadIdx.x] = loc;
  __syncthreads();
  for (int off = BDIM / 2; off > 0; off >>= 1) {
    if (threadIdx.x < off) red[threadIdx.x] += red[threadIdx.x + off];
    __syncthreads();
  }
  if (threadIdx.x == 0) atomicAdd(acc, red[0]);
}

// ---------------- finalize ----------------
__global__ void k_final(const float* acc, float* out) {
  if (blockIdx.x == 0 && threadIdx.x == 0) {
    const float n = (float)NSAMP;
    float hy_xy = acc[0] / n, hc_xy = acc[1] / n;
    float hy_yx = acc[2] / n, hc_yx = acc[3] / n;
    float te_xy = fmaxf(0.f, hy_xy - hc_xy);
    float te_yx = fmaxf(0.f, hy_yx - hc_yx);
    out[0] = te_xy;
    out[1] = te_yx;
    out[2] = te_xy - te_yx;
  }
}

// ---------------- host launch ----------------
extern "C" void kernel_launch(void* const* d_in, const int* in_sizes, int n_in,
                              void* d_out, int out_size, void* d_ws, size_t ws_size,
                              hipStream_t stream) {
  const float* src = (const float*)d_in[0];
  const float* tgt = (const float*)d_in[1];
  float* out = (float*)d_out;

  // workspace layout (all offsets 256-aligned); total ~32.03 MiB
  char* ws = (char*)d_ws;
  unsigned* ctrl = (unsigned*)ws;                                   // 8 u32
  unsigned char* s8 = (unsigned char*)(ws + 256);                   // TLEN bytes
  unsigned char* t8 = s8 + (size_t)TLEN;                            // TLEN bytes
  unsigned* jy_xy = (unsigned*)(ws + 256 + 2 * (size_t)TLEN);       // 4096 u32
  unsigned* jy_yx = jy_xy + NPAST * K_BINS;                         // 4096 u32
  unsigned* jc_xy = jy_yx + NPAST * K_BINS;                         // 2M u32
  unsigned* jc_yx = jc_xy + (size_t)NPAST * NPAST * K_BINS;         // 2M u32
  float* acc = (float*)(ctrl + 4);

  // zero all histogram counters (graph-capturable memset node)
  size_t hist_bytes = (size_t)(2 * NPAST * K_BINS + 2 * NPAST * NPAST * K_BINS)
                      * sizeof(unsigned);
  (void)hipMemsetAsync(jy_xy, 0, hist_bytes, stream);

  k_init<<<1, 1, 0, stream>>>(ctrl);

  int nvec = TLEN / 4;
  k_minmax<<<2048, BDIM, 0, stream>>>((const float4*)src, (const float4*)tgt,
                                      ctrl, nvec);
  k_disc<<<2048, BDIM, 0, stream>>>((const float4*)src, (const float4*)tgt,
                                    (uchar4*)s8, (uchar4*)t8, ctrl, nvec);

  k_hist<<<HIST_BLOCKS, BDIM, 0, stream>>>(s8, t8, jy_xy, jy_yx, jc_xy, jc_yx,
                                           NSAMP);

  k_entropy<<<8, BDIM, 0, stream>>>((const uint4*)jy_xy, NPAST, acc + 0);
  k_entropy<<<1024, BDIM, 0, stream>>>((const uint4*)jc_xy, NPAST * NPAST, acc + 1);
  k_entropy<<<8, BDIM, 0, stream>>>((const uint4*)jy_yx, NPAST, acc + 2);
  k_entropy<<<1024, BDIM, 0, stream>>>((const uint4*)jc_yx, NPAST * NPAST, acc + 3);

  k_final<<<1, 1, 0, stream>>>(acc, out);
}